// AttnDecoderRNN_51058571214881
// MI455X (gfx1250) — compile-verified
//
#include <hip/hip_runtime.h>
#include <hip/hip_bf16.h>
#include <math.h>

#define H   256
#define H2  512
#define H3  768
#define V   32000
#define L   2048
#define B   64

typedef __attribute__((ext_vector_type(2))) float v2f;
typedef __attribute__((ext_vector_type(8))) float v8f;

// workspace layout (float offsets)
#define WS_W2   0            // [2H] = 512
#define WS_BIAS 512          // [1]
#define WS_EH   516          // [B] = 64
#define WS_X    640          // [B][2H] = 32768   (x = [embedded ; context])
#define WS_E    33408        // [B][L]  = 131072  (energies)
#define WS_X2   164480       // [B][2H] = 32768   (x2 = [h_new ; context])

// d_out layout (floats): logits [B][V], h_new [B][H], attn [B][L]
#define OUT_H    (B * V)
#define OUT_ATTN (B * V + B * H)

// ---------------------------------------------------------------------------
// K1: w2 = attn_W^T @ flatten ; bias = flatten . attn_b
__global__ void k_prep(const float* __restrict__ attn_W,
                       const float* __restrict__ attn_b,
                       const float* __restrict__ flatten,
                       float* __restrict__ ws) {
    int j = threadIdx.x;             // 0..255
    float s0 = 0.f, s1 = 0.f;
    for (int i = 0; i < H; ++i) {
        float fv = flatten[i];
        s0 += attn_W[i * H2 + j] * fv;
        s1 += attn_W[i * H2 + H + j] * fv;
    }
    ws[WS_W2 + j]     = s0;
    ws[WS_W2 + H + j] = s1;

    __shared__ float red[256];
    red[j] = flatten[j] * attn_b[j];
    __syncthreads();
    for (int s = 128; s > 0; s >>= 1) {
        if (j < s) red[j] += red[j + s];
        __syncthreads();
    }
    if (j == 0) ws[WS_BIAS] = red[0];
}

// ---------------------------------------------------------------------------
// K2: embedding gather into x[:, :H] ; e_h[b] = h_b . w2[:H] + bias
__global__ void k_embed_eh(const int* __restrict__ input,
                           const float* __restrict__ hidden,
                           const float* __restrict__ emb,
                           float* __restrict__ ws) {
    int b = blockIdx.x, t = threadIdx.x;
    int tok = input[b];
    ws[WS_X + b * H2 + t] = emb[(size_t)tok * H + t];

    __shared__ float red[256];
    red[t] = hidden[b * H + t] * ws[WS_W2 + t];
    __syncthreads();
    for (int s = 128; s > 0; s >>= 1) {
        if (t < s) red[t] += red[t + s];
        __syncthreads();
    }
    if (t == 0) ws[WS_EH + b] = red[0] + ws[WS_BIAS];
}

// ---------------------------------------------------------------------------
// K3: energies E[b][l] = enc[l,b,:] . w2[H:] + e_h[b]   (wave32 per (b,l))
__global__ void k_energies(const float* __restrict__ enc,
                           float* __restrict__ ws) {
    int b    = blockIdx.y;
    int l    = blockIdx.x * 8 + (threadIdx.x >> 5);
    int lane = threadIdx.x & 31;

    const float* row = enc + ((size_t)l * B + b) * H;
    // prefetch the second half of this row's 1KB while the first half streams
    __builtin_prefetch(row + 128 + lane, 0, 1);
    const float* w2  = ws + WS_W2 + H;
    float s = 0.f;
#pragma unroll
    for (int h = lane; h < H; h += 32) s += row[h] * w2[h];
#pragma unroll
    for (int off = 16; off > 0; off >>= 1) s += __shfl_down(s, off, 32);
    if (lane == 0) ws[WS_E + b * L + l] = s + ws[WS_EH + b];
}

// ---------------------------------------------------------------------------
// K4: softmax over L per batch row -> attn region of d_out
__global__ void k_softmax(const float* __restrict__ ws,
                          float* __restrict__ attn_out) {
    int b = blockIdx.x, t = threadIdx.x;
    const float* e = ws + WS_E + b * L;
    __shared__ float red[256];

    float m = -3.402823466e38f;
    for (int l = t; l < L; l += 256) m = fmaxf(m, e[l]);
    red[t] = m; __syncthreads();
    for (int s = 128; s > 0; s >>= 1) {
        if (t < s) red[t] = fmaxf(red[t], red[t + s]);
        __syncthreads();
    }
    m = red[0]; __syncthreads();

    float sum = 0.f;
    for (int l = t; l < L; l += 256) sum += expf(e[l] - m);
    red[t] = sum; __syncthreads();
    for (int s = 128; s > 0; s >>= 1) {
        if (t < s) red[t] += red[t + s];
        __syncthreads();
    }
    float inv = 1.f / red[0];
    for (int l = t; l < L; l += 256)
        attn_out[b * L + l] = expf(e[l] - m) * inv;
}

// ---------------------------------------------------------------------------
// K5: context[b][h] = sum_l attn[b,l] * enc[l,b,h]  -> x[:,H:], x2[:,H:]
__global__ void k_context(const float* __restrict__ enc,
                          const float* __restrict__ attn,
                          float* __restrict__ ws) {
    int b = blockIdx.x, h = threadIdx.x;
    __shared__ float sa[256];
    float acc = 0.f;
    for (int lc = 0; lc < L; lc += 256) {
        __syncthreads();
        sa[h] = attn[b * L + lc + h];
        __syncthreads();
        const float* base = enc + ((size_t)lc * B + b) * H + h;
        for (int j = 0; j < 256; ++j) {
            // prefetch 8 rows (512KB/block stream) ahead of the consume point
            __builtin_prefetch(base + (size_t)(j + 8) * B * H, 0, 1);
            acc += sa[j] * base[(size_t)j * B * H];
        }
    }
    ws[WS_X  + b * H2 + H + h] = acc;
    ws[WS_X2 + b * H2 + H + h] = acc;
}

// ---------------------------------------------------------------------------
// K6: one GRU step (PyTorch gate order r,z,n); h_new -> d_out + x2[:, :H]
__global__ void k_gru(const float* __restrict__ hidden,
                      const float* __restrict__ Wih,
                      const float* __restrict__ Whh,
                      const float* __restrict__ bih,
                      const float* __restrict__ bhh,
                      float* __restrict__ ws,
                      float* __restrict__ out_h) {
    int b = blockIdx.x, h = threadIdx.x;
    __shared__ float sx[H2];
    __shared__ float sh[H];
    sx[h]     = ws[WS_X + b * H2 + h];
    sx[H + h] = ws[WS_X + b * H2 + H + h];
    sh[h]     = hidden[b * H + h];
    __syncthreads();

    float gr = bih[h], gz = bih[H + h], gn = bih[2 * H + h];
    const float* wr = Wih + (size_t)h * H2;
    const float* wz = Wih + (size_t)(H + h) * H2;
    const float* wn = Wih + (size_t)(2 * H + h) * H2;
    for (int k = 0; k < H2; ++k) {
        float xv = sx[k];
        gr += xv * wr[k]; gz += xv * wz[k]; gn += xv * wn[k];
    }
    float hr = bhh[h], hz = bhh[H + h], hn = bhh[2 * H + h];
    const float* vr = Whh + (size_t)h * H;
    const float* vz = Whh + (size_t)(H + h) * H;
    const float* vn = Whh + (size_t)(2 * H + h) * H;
    for (int k = 0; k < H; ++k) {
        float hv = sh[k];
        hr += hv * vr[k]; hz += hv * vz[k]; hn += hv * vn[k];
    }
    float r = 1.f / (1.f + expf(-(gr + hr)));
    float z = 1.f / (1.f + expf(-(gz + hz)));
    float n = tanhf(gn + r * hn);
    float hnew = (1.f - z) * n + z * sh[h];

    out_h[b * H + h]       = hnew;
    ws[WS_X2 + b * H2 + h] = hnew;
}

// ---------------------------------------------------------------------------
// K7: logits = x2 @ out_W^T + out_b  via V_WMMA_F32_16X16X4_F32.
// 256-thread blocks = 8 wave32s; each wave owns one 16-wide vocab strip and
// computes all 64 batch rows (4 M-tile accumulators share each B fragment, so
// the 65.5MB out_W stream is loaded exactly once).
// A/B lane layout: elem0 at (row/col = lane%16, K = 2*(lane/16)), elem1 at K+1.
// C/D v8f: M = 8*(lane/16) + r, N = lane%16.
__global__ void k_logits(const float* __restrict__ ws,
                         const float* __restrict__ out_W,
                         const float* __restrict__ out_b,
                         float* __restrict__ out) {
    int lane  = threadIdx.x & 31;
    int wave  = threadIdx.x >> 5;         // 0..7
    int nloc  = lane & 15;
    int n     = (blockIdx.x * 8 + wave) * 16 + nloc;
    int kb    = (lane >> 4) * 2;          // 0 or 2
    const float* x2   = ws + WS_X2;
    const float* brow = out_W + (size_t)n * H2 + kb;
    const float* a0p  = x2 + (size_t)(0 * 16 + nloc) * H2 + kb;
    const float* a1p  = x2 + (size_t)(1 * 16 + nloc) * H2 + kb;
    const float* a2p  = x2 + (size_t)(2 * 16 + nloc) * H2 + kb;
    const float* a3p  = x2 + (size_t)(3 * 16 + nloc) * H2 + kb;

    // prefetch the back half of this lane's out_W row (2KB/row stream)
    __builtin_prefetch(brow + 256, 0, 1);

    v8f acc0 = {}, acc1 = {}, acc2 = {}, acc3 = {};
    for (int k0 = 0; k0 < H2; k0 += 4) {
        v2f bf = *(const v2f*)(brow + k0);
        v2f a0 = *(const v2f*)(a0p + k0);
        v2f a1 = *(const v2f*)(a1p + k0);
        v2f a2 = *(const v2f*)(a2p + k0);
        v2f a3 = *(const v2f*)(a3p + k0);
        acc0 = __builtin_amdgcn_wmma_f32_16x16x4_f32(false, a0, false, bf, (short)0, acc0, false, false);
        acc1 = __builtin_amdgcn_wmma_f32_16x16x4_f32(false, a1, false, bf, (short)0, acc1, false, false);
        acc2 = __builtin_amdgcn_wmma_f32_16x16x4_f32(false, a2, false, bf, (short)0, acc2, false, false);
        acc3 = __builtin_amdgcn_wmma_f32_16x16x4_f32(false, a3, false, bf, (short)0, acc3, false, false);
    }

    float bias = out_b[n];
    int mrow = (lane >> 4) * 8;
#pragma unroll
    for (int r = 0; r < 8; ++r) {
        out[(size_t)( 0 + mrow + r) * V + n] = acc0[r] + bias;
        out[(size_t)(16 + mrow + r) * V + n] = acc1[r] + bias;
        out[(size_t)(32 + mrow + r) * V + n] = acc2[r] + bias;
        out[(size_t)(48 + mrow + r) * V + n] = acc3[r] + bias;
    }
}

// ---------------------------------------------------------------------------
// K8: in-place log_softmax over each [V] row of the logits region
__global__ void k_logsoftmax(float* __restrict__ out) {
    int b = blockIdx.x, t = threadIdx.x;
    float* row = out + (size_t)b * V;
    __shared__ float red[256];

    float m = -3.402823466e38f;
    for (int v = t; v < V; v += 256) m = fmaxf(m, row[v]);
    red[t] = m; __syncthreads();
    for (int s = 128; s > 0; s >>= 1) {
        if (t < s) red[t] = fmaxf(red[t], red[t + s]);
        __syncthreads();
    }
    m = red[0]; __syncthreads();

    float sum = 0.f;
    for (int v = t; v < V; v += 256) sum += expf(row[v] - m);
    red[t] = sum; __syncthreads();
    for (int s = 128; s > 0; s >>= 1) {
        if (t < s) red[t] += red[t + s];
        __syncthreads();
    }
    float lse = m + logf(red[0]);
    for (int v = t; v < V; v += 256) row[v] = row[v] - lse;
}

// ---------------------------------------------------------------------------
extern "C" void kernel_launch(void* const* d_in, const int* in_sizes, int n_in,
                              void* d_out, int out_size, void* d_ws, size_t ws_size,
                              hipStream_t stream) {
    const int*   input   = (const int*)d_in[0];
    const float* hidden  = (const float*)d_in[1];
    const float* enc     = (const float*)d_in[2];
    // d_in[3] = use_cuda (ignored)
    const float* emb     = (const float*)d_in[4];
    const float* attn_W  = (const float*)d_in[5];
    const float* attn_b  = (const float*)d_in[6];
    const float* flatten = (const float*)d_in[7];
    const float* Wih     = (const float*)d_in[8];
    const float* Whh     = (const float*)d_in[9];
    const float* bih     = (const float*)d_in[10];
    const float* bhh     = (const float*)d_in[11];
    const float* outW    = (const float*)d_in[12];
    const float* outb    = (const float*)d_in[13];

    float* out = (float*)d_out;
    float* ws  = (float*)d_ws;

    k_prep      <<<1,              256, 0, stream>>>(attn_W, attn_b, flatten, ws);
    k_embed_eh  <<<B,              256, 0, stream>>>(input, hidden, emb, ws);
    k_energies  <<<dim3(L / 8, B), 256, 0, stream>>>(enc, ws);
    k_softmax   <<<B,              256, 0, stream>>>(ws, out + OUT_ATTN);
    k_context   <<<B,              256, 0, stream>>>(enc, out + OUT_ATTN, ws);
    k_gru       <<<B,              256, 0, stream>>>(hidden, Wih, Whh, bih, bhh, ws, out + OUT_H);
    k_logits    <<<V / (16 * 8),   256, 0, stream>>>(ws, outW, outb, out);
    k_logsoftmax<<<B,              256, 0, stream>>>(out);
}